// LSEPLoss_29532195127620
// MI455X (gfx1250) — compile-verified
//
#include <hip/hip_runtime.h>
#include <hip/hip_bf16.h>
#include <cmath>

// ---------------------------------------------------------------------------
// LSEP loss:  out = log1p( sum_i  (sum_{t==0} exp(x)) * (sum_{t>0} exp(-x)) )
// B = 8192 rows, L = 5000 cols.  Pure streaming reduction: 328 MB read once.
// HBM floor ~14us @ 23.3 TB/s  ->  optimize for B128 nontemporal loads.
// ---------------------------------------------------------------------------

typedef __attribute__((ext_vector_type(4))) float v4f;
typedef __attribute__((ext_vector_type(4))) int   v4i;
typedef __attribute__((ext_vector_type(2))) float v2f;
typedef __attribute__((ext_vector_type(8))) float v8f;

static constexpr int kL     = 5000;   // columns per row
static constexpr int kLvec  = kL / 4; // 1250 float4 per row (row stride 20000B, 16B aligned)
static constexpr int kBlock = 256;    // 8 waves (wave32)

// --------------------------- kernel 1: per-row -----------------------------
// One workgroup per row. Each thread strides over the row's float4/int4
// vectors with nontemporal B128 loads, accumulating neg_sum (t==0: exp(x))
// and pos_sum (t>0: exp(-x)) with a single exp per element (sign flip).
__global__ __launch_bounds__(kBlock)
void lsep_row_kernel(const float* __restrict__ x,
                     const int*   __restrict__ t,
                     float*       __restrict__ rowprod) {
    const int row = blockIdx.x;
    const int tid = threadIdx.x;

    const v4f* __restrict__ xr = reinterpret_cast<const v4f*>(x + (size_t)row * kL);
    const v4i* __restrict__ tr = reinterpret_cast<const v4i*>(t + (size_t)row * kL);

    float negAcc = 0.0f;
    float posAcc = 0.0f;

    for (int vi = tid; vi < kLvec; vi += kBlock) {
        // Single-use streaming data: nontemporal so we don't thrash the 192MB L2.
        v4f xv = __builtin_nontemporal_load(xr + vi);
        v4i tv = __builtin_nontemporal_load(tr + vi);
#pragma unroll
        for (int j = 0; j < 4; ++j) {
            const float xe  = xv[j];
            const bool  pos = (tv[j] != 0);          // target in {0,1}
            const float e   = __expf(pos ? -xe : xe); // one transcendental/elem
            negAcc += pos ? 0.0f : e;
            posAcc += pos ? e : 0.0f;
        }
    }

    __shared__ float sneg[kBlock];
    __shared__ float spos[kBlock];
    sneg[tid] = negAcc;
    spos[tid] = posAcc;
    __syncthreads();

#pragma unroll
    for (int s = kBlock / 2; s > 0; s >>= 1) {
        if (tid < s) {
            sneg[tid] += sneg[tid + s];
            spos[tid] += spos[tid + s];
        }
        __syncthreads();
    }

    if (tid == 0) {
        // sum_{p,n} exp(x_n - x_p) factors into neg_sum * pos_sum per row
        rowprod[row] = sneg[0] * spos[0];
    }
}

// --------------------------- kernel 2: finalize ----------------------------
// Single block, deterministic tree sum of the 8192 per-row products, then
// log1p. Cross-lane reduction of each wave's 32 partials is done with
// V_WMMA_F32_16X16X4_F32 (A = partials with a1=0, B = ones):
//   D[m,n] = A[m,0]+A[m,1]+A[m,2]+A[m,3] = v_m + v_{m+16}
// so sum(d[0..7]) per lane + one shfl_xor(16) yields the full 32-lane sum
// in fp32 (f16 WMMA would overflow: row products ~1e7).
__global__ __launch_bounds__(kBlock)
void lsep_finalize_kernel(const float* __restrict__ rowprod,
                          float*       __restrict__ out,
                          int n) {
    const int tid  = threadIdx.x;
    const int lane = tid & 31;
    const int wave = tid >> 5;

    float s = 0.0f;
    for (int i = tid; i < n; i += kBlock) s += rowprod[i];

    float waveSum;
#if __has_builtin(__builtin_amdgcn_wmma_f32_16x16x4_f32)
    {
        v2f a; a[0] = s;    a[1] = 0.0f;   // A 16x4: lane m -> A[m%16, {0,1} or {2,3}]
        v2f b; b[0] = 1.0f; b[1] = 1.0f;   // B 4x16: all ones
        v8f c = {};
        c = __builtin_amdgcn_wmma_f32_16x16x4_f32(
                /*neg_a=*/false, a, /*neg_b=*/false, b,
                /*c_mod=*/(short)0, c, /*reuse_a=*/false, /*reuse_b=*/false);
        // lanes 0-15 hold rows M=0..7 in c[0..7]; lanes 16-31 hold M=8..15
        float p = c[0] + c[1] + c[2] + c[3] + c[4] + c[5] + c[6] + c[7];
        waveSum = p + __shfl_xor(p, 16, 32);  // add the other half's rows
    }
#else
    {
        float p = s;
#pragma unroll
        for (int m = 16; m > 0; m >>= 1) p += __shfl_xor(p, m, 32);
        waveSum = p;
    }
#endif

    __shared__ float wsum[kBlock / 32];
    if (lane == 0) wsum[wave] = waveSum;
    __syncthreads();

    if (tid == 0) {
        float total = 0.0f;
#pragma unroll
        for (int w = 0; w < kBlock / 32; ++w) total += wsum[w];
        out[0] = log1pf(total);
    }
}

// ---------------------------------------------------------------------------
extern "C" void kernel_launch(void* const* d_in, const int* in_sizes, int n_in,
                              void* d_out, int out_size, void* d_ws, size_t ws_size,
                              hipStream_t stream) {
    const float* x = (const float*)d_in[0];
    const int*   t = (const int*)d_in[1];
    float*       out = (float*)d_out;
    float*       rowprod = (float*)d_ws;        // rows * 4 bytes of scratch

    const int rows = in_sizes[0] / kL;          // 8192

    lsep_row_kernel<<<rows, kBlock, 0, stream>>>(x, t, rowprod);
    lsep_finalize_kernel<<<1, kBlock, 0, stream>>>(rowprod, out, rows);
}